// Encoder_72095321030804
// MI455X (gfx1250) — compile-verified
//
#include <hip/hip_runtime.h>

typedef __attribute__((ext_vector_type(2))) float v2f;
typedef __attribute__((ext_vector_type(8))) float v8f;

#define T_STEPS 256
#define BATCH   256
#define EMB     256
#define HID     512
#define G4H     (4 * HID)

__device__ __forceinline__ float sigmoidf_(float x) {
    return 1.0f / (1.0f + __expf(-x));
}

__global__ void zero_kernel(float* __restrict__ p, int n) {
    int i = blockIdx.x * blockDim.x + threadIdx.x;
    if (i < n) p[i] = 0.0f;
}

// ---- K-chunk helpers: one chunk = 16 K values = 4 fp32 WMMA steps ----
// arow/w* pointers are pre-offset by koff (per-lane K sub-offset).
__device__ __forceinline__ void load_chunk(v2f a[4], v2f bb[4][4],
    const float* __restrict__ arow,
    const float* __restrict__ w0, const float* __restrict__ w1,
    const float* __restrict__ w2, const float* __restrict__ w3, int k)
{
#pragma unroll
    for (int u = 0; u < 4; ++u) {
        a[u]     = *(const v2f*)(arow + k + u * 4);
        bb[0][u] = *(const v2f*)(w0 + k + u * 4);
        bb[1][u] = *(const v2f*)(w1 + k + u * 4);
        bb[2][u] = *(const v2f*)(w2 + k + u * 4);
        bb[3][u] = *(const v2f*)(w3 + k + u * 4);
    }
}

__device__ __forceinline__ void wmma_chunk(v8f acc[4], const v2f a[4],
                                           const v2f bb[4][4])
{
#pragma unroll
    for (int u = 0; u < 4; ++u) {
#pragma unroll
        for (int q = 0; q < 4; ++q) {
            acc[q] = __builtin_amdgcn_wmma_f32_16x16x4_f32(
                false, a[u], false, bb[q][u], (short)0, acc[q], false, false);
        }
    }
}

// Double-buffered GEMM over K (K multiple of 32): loads for chunk i+1 are in
// flight while the 16 WMMAs of chunk i issue -> partial loadcnt waits.
__device__ __forceinline__ void gemm_K(v8f acc[4],
    const float* __restrict__ arow,
    const float* __restrict__ w0, const float* __restrict__ w1,
    const float* __restrict__ w2, const float* __restrict__ w3, int K)
{
    v2f a0[4], b0[4][4], a1[4], b1[4][4];
    const int nch = K >> 4;  // #chunks, even
    load_chunk(a0, b0, arow, w0, w1, w2, w3, 0);
    int c = 0;
    for (; c + 2 < nch; c += 2) {
        load_chunk(a1, b1, arow, w0, w1, w2, w3, (c + 1) << 4);
        wmma_chunk(acc, a0, b0);
        load_chunk(a0, b0, arow, w0, w1, w2, w3, (c + 2) << 4);
        wmma_chunk(acc, a1, b1);
    }
    load_chunk(a1, b1, arow, w0, w1, w2, w3, (nch - 1) << 4);
    wmma_chunk(acc, a0, b0);
    wmma_chunk(acc, a1, b1);
}

// ---- Parallel input projection: G0[r, g] = emb[x[r]] @ W_ih^T + b_ih + b_hh
// over all r in [0, T*B). Fully parallel, off the recurrent critical path.
__global__ __launch_bounds__(128) void input_proj_kernel(
    const int*   __restrict__ x_all,  // [T*B]
    const float* __restrict__ emb,    // [V, E]
    const float* __restrict__ W_ih,   // [4H, E]
    const float* __restrict__ b_ih,   // [4H]
    const float* __restrict__ b_hh,   // [4H]
    float*       __restrict__ G0)     // [T*B, 4H]
{
    const int lane = threadIdx.x & 31;
    const int wave = threadIdx.x >> 5;
    const int tile = blockIdx.x * (blockDim.x >> 5) + wave; // 0..131071
    const int r0 = (tile >> 5) * 16;   // row tile (4096 tiles)
    const int h0 = (tile & 31) * 16;   // h-col tile (32 tiles)

    const int n    = lane & 15;
    const int hi   = lane >> 4;
    const int koff = hi * 2;

    const int rA  = r0 + n;
    const int idx = x_all[rA];

    v8f acc[4];
#pragma unroll
    for (int q = 0; q < 4; ++q) {
        const int g = q * HID + h0 + n;
        const float bv = b_ih[g] + b_hh[g];
        v8f t;
#pragma unroll
        for (int j = 0; j < 8; ++j) t[j] = bv;
        acc[q] = t;
    }

    const float* arow = emb + idx * EMB + koff;
    const float* w0 = W_ih + (0 * HID + h0 + n) * EMB + koff;
    const float* w1 = W_ih + (1 * HID + h0 + n) * EMB + koff;
    const float* w2 = W_ih + (2 * HID + h0 + n) * EMB + koff;
    const float* w3 = W_ih + (3 * HID + h0 + n) * EMB + koff;
    gemm_K(acc, arow, w0, w1, w2, w3, EMB);

#pragma unroll
    for (int q = 0; q < 4; ++q) {
#pragma unroll
        for (int j = 0; j < 8; ++j) {
            const int row = r0 + (hi ? (j + 8) : j);
            G0[row * G4H + q * HID + h0 + n] = acc[q][j];
        }
    }
}

// ---- One LSTM timestep ----
__global__ __launch_bounds__(128) void lstm_step_kernel(
    const int*   __restrict__ x_t,       // [B] token ids (fused path)
    const float* __restrict__ emb,       // [V, E]
    const float* __restrict__ W_ih,      // [4H, E]
    const float* __restrict__ W_hh,      // [4H, H]
    const float* __restrict__ b_ih,      // [4H]
    const float* __restrict__ b_hh,      // [4H]
    const float* __restrict__ gate_init, // [B, 4H] precomputed, or nullptr
    const float* __restrict__ h_prev,    // [B, H] or nullptr (t == 0)
    float*       __restrict__ h_out,     // [B, H] (outs[t])
    float*       __restrict__ c_buf,     // [B, H] running cell state
    float*       __restrict__ h_final,   // [B, H] final-h region
    int is_last)
{
    const int lane = threadIdx.x & 31;
    const int wave = threadIdx.x >> 5;
    const int tile = blockIdx.x * (blockDim.x >> 5) + wave; // 0..511
    const int b0 = (tile & 15) * 16;
    const int h0 = (tile >> 4) * 16;

    const int n    = lane & 15;
    const int hi   = lane >> 4;
    const int koff = hi * 2;
    const int bA   = b0 + n;

    v8f acc[4];
    if (gate_init != nullptr) {
        // Input projection + biases already computed.
#pragma unroll
        for (int q = 0; q < 4; ++q) {
            v8f t;
#pragma unroll
            for (int j = 0; j < 8; ++j) {
                const int row = b0 + (hi ? (j + 8) : j);
                t[j] = gate_init[row * G4H + q * HID + h0 + n];
            }
            acc[q] = t;
        }
    } else {
        // Fused path: biases then embedding projection.
#pragma unroll
        for (int q = 0; q < 4; ++q) {
            const int g = q * HID + h0 + n;
            const float bv = b_ih[g] + b_hh[g];
            v8f t;
#pragma unroll
            for (int j = 0; j < 8; ++j) t[j] = bv;
            acc[q] = t;
        }
        const int idx = x_t[bA];
        const float* arow = emb + idx * EMB + koff;
        const float* w0 = W_ih + (0 * HID + h0 + n) * EMB + koff;
        const float* w1 = W_ih + (1 * HID + h0 + n) * EMB + koff;
        const float* w2 = W_ih + (2 * HID + h0 + n) * EMB + koff;
        const float* w3 = W_ih + (3 * HID + h0 + n) * EMB + koff;
        gemm_K(acc, arow, w0, w1, w2, w3, EMB);
    }

    // Recurrent projection (skip at t==0: h == 0 contributes nothing).
    if (h_prev != nullptr) {
        const float* arow = h_prev + bA * HID + koff;
        const float* w0 = W_hh + (0 * HID + h0 + n) * HID + koff;
        const float* w1 = W_hh + (1 * HID + h0 + n) * HID + koff;
        const float* w2 = W_hh + (2 * HID + h0 + n) * HID + koff;
        const float* w3 = W_hh + (3 * HID + h0 + n) * HID + koff;
        gemm_K(acc, arow, w0, w1, w2, w3, HID);
    }

    // LSTM cell update: C/D layout -> element j is row (j or j+8).
#pragma unroll
    for (int j = 0; j < 8; ++j) {
        const int brow = b0 + (hi ? (j + 8) : j);
        const int off  = brow * HID + h0 + n;
        const float c_old = c_buf[off];
        const float iv = sigmoidf_(acc[0][j]);
        const float fv = sigmoidf_(acc[1][j]);
        const float gv = tanhf(acc[2][j]);
        const float ov = sigmoidf_(acc[3][j]);
        const float cn = fv * c_old + iv * gv;
        const float hn = ov * tanhf(cn);
        c_buf[off] = cn;
        h_out[off] = hn;
        if (is_last) h_final[off] = hn;
    }
}

extern "C" void kernel_launch(void* const* d_in, const int* in_sizes, int n_in,
                              void* d_out, int out_size, void* d_ws, size_t ws_size,
                              hipStream_t stream) {
    (void)in_sizes; (void)n_in; (void)out_size;
    const int*   x_seq = (const int*)d_in[0];    // [T, B]
    const float* emb   = (const float*)d_in[1];  // [V, E]
    const float* W_ih  = (const float*)d_in[2];  // [4H, E]
    const float* W_hh  = (const float*)d_in[3];  // [4H, H]
    const float* b_ih  = (const float*)d_in[4];  // [4H]
    const float* b_hh  = (const float*)d_in[5];  // [4H]

    float* out  = (float*)d_out;
    float* outs = out;                                  // [T, B, H]
    float* hfin = out + (size_t)T_STEPS * BATCH * HID;  // [B, H]
    float* cbuf = hfin + (size_t)BATCH * HID;           // [B, H] (final c)

    // Zero the running cell state (d_out is poisoned by the harness).
    const int cn = BATCH * HID;
    zero_kernel<<<(cn + 255) / 256, 256, 0, stream>>>(cbuf, cn);

    // If workspace is big enough, precompute the input projection for all
    // timesteps in one fully-parallel GEMM (takes E off the serial chain).
    const size_t g0_bytes = (size_t)T_STEPS * BATCH * G4H * sizeof(float);
    float* G0 = nullptr;
    if (ws_size >= g0_bytes) {
        G0 = (float*)d_ws;
        // (T*B/16) * 32 tiles = 131072 waves; 4 waves/block -> 32768 blocks.
        input_proj_kernel<<<32768, 128, 0, stream>>>(x_seq, emb, W_ih,
                                                     b_ih, b_hh, G0);
    }

    // 512 wave-tiles per step, 4 waves per 128-thread block -> 128 blocks.
    for (int t = 0; t < T_STEPS; ++t) {
        const float* hprev = (t == 0) ? nullptr
                                      : (outs + (size_t)(t - 1) * BATCH * HID);
        const float* gi = G0 ? (G0 + (size_t)t * BATCH * G4H) : nullptr;
        lstm_step_kernel<<<128, 128, 0, stream>>>(
            x_seq + (size_t)t * BATCH, emb, W_ih, W_hh, b_ih, b_hh,
            gi, hprev,
            outs + (size_t)t * BATCH * HID,
            cbuf, hfin,
            (t == T_STEPS - 1) ? 1 : 0);
    }
}